// CropRoi_11312943858080
// MI455X (gfx1250) — compile-verified
//
#include <hip/hip_runtime.h>
#include <float.h>
#include <stdint.h>

// ---------------------------------------------------------------------------
// CropRoi adaptive 3D max-pool (R=7) for f:(4,64,32,32,32) fp32, N=96 proposals.
// CDNA5 path: TDM TENSOR_LOAD_TO_LDS (inline asm, VIMAGE encoding: 4 SGPR
// groups) stages the cropped (Lx,Ly,Lz,4ch) 4D tile into LDS, double-buffered
// with s_wait_tensorcnt; 8 wave32 pool from LDS.
// ---------------------------------------------------------------------------

typedef __attribute__((ext_vector_type(4))) unsigned int v4u_;
typedef __attribute__((ext_vector_type(8))) int          v8i_;
typedef __attribute__((ext_vector_type(4))) int          v4i_;

#define C_TOT    64
#define FS       32          // feature spatial extent
#define RB       7           // pooled bins per axis
#define CH       4           // channels per TDM stage
#define TILE_MAX 2744        // 14^3 upper bound on crop volume (side<=48 -> L<=14)
#define NTHREADS 256

#if __has_builtin(__builtin_amdgcn_s_wait_tensorcnt)
#define WAIT_TENSORCNT(n) __builtin_amdgcn_s_wait_tensorcnt((short)(n))
#else
#define WAIT_TENSORCNT(n) asm volatile("s_wait_tensorcnt %0" :: "i"(n) : "memory")
#endif

// Issue one TDM descriptor: 4D tile (Lx, Ly, Lz, CH channels) of the 32^3 x 64ch
// tensor starting at gptr, into LDS byte offset lds_byte_off. Bit layout per
// CDNA5 ISA. Inline asm bypasses the clang builtin
// (whose arity differs across toolchains); groups land in aligned SGPR tuples.
__device__ __forceinline__ void tdm_load_tile(const float* gptr,
                                              unsigned lds_byte_off,
                                              int Lx, int Ly, int Lz)
{
  unsigned long long ga = (unsigned long long)(uintptr_t)gptr;
  v4u_ g0;
  g0[0] = 1u;                                        // count=1, user D#, no gather
  g0[1] = lds_byte_off;                              // lds_addr (bytes)
  g0[2] = (unsigned)ga;                              // global_addr[31:0]
  g0[3] = (unsigned)((ga >> 32) & 0x01FFFFFFu)       // global_addr[56:32]
        | (2u << 30);                                // type = 2 ("image")
  v8i_ g1;
  g1[0] = (int)(2u << 16);                           // wg_mask=0 | data_size=2 (4B)
  g1[1] = (int)((unsigned)FS << 16);                 // abar=0 | tensor_dim0[15:0]=32
  g1[2] = (int)((unsigned)FS << 16);                 // dim0[31:16]=0 | dim1[15:0]=32
  g1[3] = (int)((unsigned)Lx << 16);                 // dim1[31:16]=0 | tile_dim0=Lx
  g1[4] = (int)((unsigned)Ly | ((unsigned)Lz << 16));// tile_dim1=Ly | tile_dim2=Lz
  g1[5] = FS;                                        // tensor_dim0_stride[31:0] = 32
  g1[6] = (int)(((unsigned)(FS * FS) & 0xFFFFu) << 16); // s0[47:32]=0 | s1[15:0]=1024
  g1[7] = (int)((unsigned)(FS * FS) >> 16);          // tensor_dim1_stride[47:16] = 0
  v4i_ g2;
  g2[0] = FS;                                        // tensor_dim2 = 32 (z extent)
  g2[1] = C_TOT;                                     // tensor_dim3 = 64 (channels)
  g2[2] = FS * FS * FS;                              // tensor_dim2_stride = 32768
  g2[3] = (int)((unsigned)CH << 16);                 // s2[47:32]=0 | tile_dim3 = CH
  v4i_ g3 = (v4i_)0;                                 // no 5th dimension
  asm volatile("tensor_load_to_lds %0, %1, %2, %3"
               :
               : "s"(g0), "s"(g1), "s"(g2), "s"(g3)
               : "memory");
}

__global__ __launch_bounds__(NTHREADS)
void crop_pool_kernel(const float* __restrict__ f,
                      const float* __restrict__ proposals,
                      float* __restrict__ out)
{
  __shared__ float sbuf[2][CH * TILE_MAX];
  __shared__ int   sbnd[3][RB][2];   // per-axis bin [start,end) relative to c0

  const int n = blockIdx.x;
  const float* p = proposals + (size_t)n * 8;

  // Uniform per-proposal crop bounds (match jnp fp32 math; /4 and /2 are exact).
  const int b = (int)p[0];
  int c0[3], c1[3];
#pragma unroll
  for (int ax = 0; ax < 3; ++ax) {
    float ctr = p[2 + ax], sd = p[5 + ax];
    int a = (int)floorf((ctr - 0.5f * sd) * 0.25f);
    int d = (int)ceilf((ctr + 0.5f * sd) * 0.25f);
    a = a > 0 ? a : 0;
    d = d < FS ? d : FS;
    if (d < a) d = a;
    c0[ax] = a; c1[ax] = d;
  }
  const int Lz = c1[0] - c0[0], Ly = c1[1] - c0[1], Lx = c1[2] - c0[2];
  const int Lv = Lx * Ly * Lz;

  // Adaptive-pool bin boundaries (empty bins -> start==end -> -FLT_MAX).
  if (threadIdx.x < 3 * RB) {
    int ax = threadIdx.x / RB, i = threadIdx.x % RB;
    int L = c1[ax] - c0[ax];
    sbnd[ax][i][0] = (i * L) / RB;
    sbnd[ax][i][1] = ((i + 1) * L + (RB - 1)) / RB;
  }

  const float* base = f + ((size_t)b * C_TOT) * (FS * FS * FS)
                        + (size_t)c0[0] * (FS * FS)
                        + (size_t)c0[1] * FS
                        + (size_t)c0[2];

  // Prefetch stage 0 (wave 0 issues; TDM ignores EXEC, tracked on wave 0's TENSORcnt).
  if (threadIdx.x == 0)
    tdm_load_tile(base, (unsigned)(uintptr_t)&sbuf[0][0], Lx, Ly, Lz);
  __syncthreads();   // publish sbnd (and order vs. stage-0 consume below)

  const int stages = C_TOT / CH;
  for (int cc = 0; cc < stages; ++cc) {
    if (threadIdx.x == 0) {
      if (cc + 1 < stages) {
        tdm_load_tile(base + (size_t)(cc + 1) * CH * (FS * FS * FS),
                      (unsigned)(uintptr_t)&sbuf[(cc + 1) & 1][0], Lx, Ly, Lz);
        WAIT_TENSORCNT(1);   // stage cc complete (TDM ops complete in order)
      } else {
        WAIT_TENSORCNT(0);
      }
    }
    __syncthreads();   // tile in LDS visible to all 8 waves

    const float* tile = &sbuf[cc & 1][0];
    for (int o = threadIdx.x; o < CH * (RB * RB * RB); o += NTHREADS) {
      const int cr  = o / (RB * RB * RB);
      const int rem = o % (RB * RB * RB);
      const int i = rem / (RB * RB);
      const int r2 = rem % (RB * RB);
      const int j = r2 / RB;
      const int k = r2 % RB;
      const int zs = sbnd[0][i][0], ze = sbnd[0][i][1];
      const int ys = sbnd[1][j][0], ye = sbnd[1][j][1];
      const int xs = sbnd[2][k][0], xe = sbnd[2][k][1];
      float v = -FLT_MAX;   // == jnp.finfo(float32).min fill for empty bins
      const float* cb = tile + cr * Lv;
      for (int z = zs; z < ze; ++z)
        for (int y = ys; y < ye; ++y) {
          const float* row = cb + ((size_t)z * Ly + y) * Lx;
          for (int x = xs; x < xe; ++x)
            v = fmaxf(v, row[x]);
        }
      out[((size_t)n * C_TOT + (size_t)cc * CH + cr) * (RB * RB * RB) + rem] = v;
    }
    __syncthreads();   // all waves done reading this buffer before it is reloaded
  }
}

extern "C" void kernel_launch(void* const* d_in, const int* in_sizes, int n_in,
                              void* d_out, int out_size, void* d_ws, size_t ws_size,
                              hipStream_t stream) {
  (void)n_in; (void)out_size; (void)d_ws; (void)ws_size;
  const float* f         = (const float*)d_in[0];   // (4,64,32,32,32) fp32
  const float* proposals = (const float*)d_in[2];   // (N,8) fp32
  float* out             = (float*)d_out;           // (N,64,7,7,7) fp32
  const int N = in_sizes[2] / 8;                    // 96
  crop_pool_kernel<<<dim3(N), dim3(NTHREADS), 0, stream>>>(f, proposals, out);
}